// MorseLoss_quad_mesh_33586644255295
// MI455X (gfx1250) — compile-verified
//
#include <hip/hip_runtime.h>
#include <math.h>

// ---------------------------------------------------------------------------
// MorseLoss quad-mesh loss for MI455X (gfx1250, wave32).
// Memory-bound reduction (~140 MB traffic, ~6-8us floor @ 23.3 TB/s HBM).
// Cross-lane reduction implemented with V_WMMA_F32_16X16X4_F32 (exact f32):
//   A[m][0]=acc[m], A[m][2]=acc[m+16], B=ones  =>  D[m][*]=acc[m]+acc[m+16]
// ---------------------------------------------------------------------------

typedef float v2f __attribute__((ext_vector_type(2)));
typedef float v8f __attribute__((ext_vector_type(8)));

#define BLOCK 256
#define NWAVES (BLOCK / 32)

// Exact full-wave (32 lane) f32 sum via the matrix unit. Requires EXEC all-1s
// (call from convergent code only). Returns total in every lane.
__device__ __forceinline__ float wave_sum_wmma(float x) {
    v2f a; a[0] = x;    a[1] = 0.0f;   // lane L<16: A[L][0]=x, A[L][1]=0
                                        // lane L>=16: A[L-16][2]=x, A[L-16][3]=0
    v2f b; b[0] = 1.0f; b[1] = 1.0f;   // B = ones (4x16)
    v8f c = {0.f, 0.f, 0.f, 0.f, 0.f, 0.f, 0.f, 0.f};
    v8f d = __builtin_amdgcn_wmma_f32_16x16x4_f32(
        /*neg_a=*/false, a, /*neg_b=*/false, b,
        /*c_mod=*/(short)0, c, /*reuse_a=*/false, /*reuse_b=*/false);
    // lane<16 holds D[0..7][lane] = acc[0..7]+acc[16..23]
    // lane>=16 holds D[8..15][lane-16] = acc[8..15]+acc[24..31]
    float s = d[0] + d[1] + d[2] + d[3] + d[4] + d[5] + d[6] + d[7];
    s += __shfl_xor(s, 16);
    return s;
}

// Cross-field directions at vertex n (normalized va, vb).
__device__ __forceinline__ void cross_dirs(int n,
                                           const float* __restrict__ th,
                                           const float* __restrict__ u,
                                           const float* __restrict__ v,
                                           float va[3], float vb[3]) {
    float t = th[n];
    float st, ct;
    sincosf(t, &st, &ct);
    float ux = u[3 * n], uy = u[3 * n + 1], uz = u[3 * n + 2];
    float vx = v[3 * n], vy = v[3 * n + 1], vz = v[3 * n + 2];
    float ax = ux * ct + vx * st, ay = uy * ct + vy * st, az = uz * ct + vz * st;
    float ia = 1.0f / (sqrtf(ax * ax + ay * ay + az * az) + 1e-12f);
    va[0] = ax * ia; va[1] = ay * ia; va[2] = az * ia;
    float bx = -ux * st + vx * ct, by = -uy * st + vy * ct, bz = -uz * st + vz * ct;
    float ib = 1.0f / (sqrtf(bx * bx + by * by + bz * bz) + 1e-12f);
    vb[0] = bx * ib; vb[1] = by * ib; vb[2] = bz * ib;
}

__global__ void zero_acc_kernel(float* __restrict__ acc) {
    if (threadIdx.x < 8) acc[threadIdx.x] = 0.0f;
}

// acc slots: 0=sdf 1=inter 2=eikonal 3=morse 4=theta_hessian 5=neighbors
__global__ void __launch_bounds__(BLOCK)
vertex_kernel(const float* __restrict__ pred,   // [N]
              const float* __restrict__ grad,   // [N,3]
              const float* __restrict__ H,      // [N,3,3]
              const float* __restrict__ ngt,    // [N,3]
              const float* __restrict__ theta,  // [N]
              const float* __restrict__ u,      // [N,3]
              const float* __restrict__ v,      // [N,3]
              float* __restrict__ acc, int N) {
    float a_sdf = 0.f, a_eik = 0.f, a_morse = 0.f, a_th = 0.f;
    for (int n = blockIdx.x * blockDim.x + threadIdx.x; n < N;
         n += gridDim.x * blockDim.x) {
        a_sdf += fabsf(pred[n]);

        float gx = grad[3 * n], gy = grad[3 * n + 1], gz = grad[3 * n + 2];
        a_eik += fabsf(sqrtf(gx * gx + gy * gy + gz * gz) - 1.0f);

        const float* Hn = H + 9 * n;
        float h00 = Hn[0], h01 = Hn[1], h02 = Hn[2];
        float h10 = Hn[3], h11 = Hn[4], h12 = Hn[5];
        float h20 = Hn[6], h21 = Hn[7], h22 = Hn[8];

        // morse: | (n_gt^T H)_e | summed over e
        float n0 = ngt[3 * n], n1 = ngt[3 * n + 1], n2 = ngt[3 * n + 2];
        a_morse += fabsf(n0 * h00 + n1 * h10 + n2 * h20)
                 + fabsf(n0 * h01 + n1 * h11 + n2 * h21)
                 + fabsf(n0 * h02 + n1 * h12 + n2 * h22);

        // cross-field hessian alignment
        float va[3], vb[3];
        cross_dirs(n, theta, u, v, va, vb);

        float pA0 = va[0] * h00 + va[1] * h10 + va[2] * h20;
        float pA1 = va[0] * h01 + va[1] * h11 + va[2] * h21;
        float pA2 = va[0] * h02 + va[1] * h12 + va[2] * h22;
        // cross(vaH, va)
        a_th += fabsf(pA1 * va[2] - pA2 * va[1])
              + fabsf(pA2 * va[0] - pA0 * va[2])
              + fabsf(pA0 * va[1] - pA1 * va[0]);

        float pB0 = vb[0] * h00 + vb[1] * h10 + vb[2] * h20;
        float pB1 = vb[0] * h01 + vb[1] * h11 + vb[2] * h21;
        float pB2 = vb[0] * h02 + vb[1] * h12 + vb[2] * h22;
        a_th += fabsf(pB1 * vb[2] - pB2 * vb[1])
              + fabsf(pB2 * vb[0] - pB0 * vb[2])
              + fabsf(pB0 * vb[1] - pB1 * vb[0]);
    }

    __shared__ float smem[4][NWAVES];
    int lane = threadIdx.x & 31, wave = threadIdx.x >> 5;
    float t0 = wave_sum_wmma(a_sdf);
    float t1 = wave_sum_wmma(a_eik);
    float t2 = wave_sum_wmma(a_morse);
    float t3 = wave_sum_wmma(a_th);
    if (lane == 0) {
        smem[0][wave] = t0; smem[1][wave] = t1;
        smem[2][wave] = t2; smem[3][wave] = t3;
    }
    __syncthreads();
    if (threadIdx.x < 4) {
        float s = 0.f;
        for (int w = 0; w < NWAVES; ++w) s += smem[threadIdx.x][w];
        const int slot[4] = {0, 2, 3, 4};
        atomicAdd(&acc[slot[threadIdx.x]], s);
    }
}

__global__ void __launch_bounds__(BLOCK)
nonmnfld_kernel(const float* __restrict__ pred,  // [M]
                const float* __restrict__ grad,  // [M,3]
                float* __restrict__ acc, int M) {
    float a_int = 0.f, a_eik = 0.f;
    for (int m = blockIdx.x * blockDim.x + threadIdx.x; m < M;
         m += gridDim.x * blockDim.x) {
        a_int += __expf(-100.0f * fabsf(pred[m]));
        float gx = grad[3 * m], gy = grad[3 * m + 1], gz = grad[3 * m + 2];
        a_eik += fabsf(sqrtf(gx * gx + gy * gy + gz * gz) - 1.0f);
    }
    __shared__ float smem[2][NWAVES];
    int lane = threadIdx.x & 31, wave = threadIdx.x >> 5;
    float t0 = wave_sum_wmma(a_int);
    float t1 = wave_sum_wmma(a_eik);
    if (lane == 0) { smem[0][wave] = t0; smem[1][wave] = t1; }
    __syncthreads();
    if (threadIdx.x < 2) {
        float s = 0.f;
        for (int w = 0; w < NWAVES; ++w) s += smem[threadIdx.x][w];
        const int slot[2] = {1, 2};
        atomicAdd(&acc[slot[threadIdx.x]], s);
    }
}

__global__ void __launch_bounds__(BLOCK)
neighbor_kernel(const float* __restrict__ theta,
                const float* __restrict__ u,
                const float* __restrict__ v,
                const float* __restrict__ rot,  // [N,K,3,3]
                const int*   __restrict__ nbr,  // [N,K]
                float* __restrict__ acc, int N, int K) {
    float a_nb = 0.f;
    for (int n = blockIdx.x * blockDim.x + threadIdx.x; n < N;
         n += gridDim.x * blockDim.x) {
        float va[3], vb[3];
        cross_dirs(n, theta, u, v, va, vb);
        for (int k = 0; k < K; ++k) {
            int j = nbr[n * K + k];
            float na[3], nb_[3];
            cross_dirs(j, theta, u, v, na, nb_);  // u/v/theta L2-resident
            const float* R = rot + 9 * (n * K + k);
            float ax = R[0] * na[0] + R[1] * na[1] + R[2] * na[2];
            float ay = R[3] * na[0] + R[4] * na[1] + R[5] * na[2];
            float az = R[6] * na[0] + R[7] * na[1] + R[8] * na[2];
            float bx = R[0] * nb_[0] + R[1] * nb_[1] + R[2] * nb_[2];
            float by = R[3] * nb_[0] + R[4] * nb_[1] + R[5] * nb_[2];
            float bz = R[6] * nb_[0] + R[7] * nb_[1] + R[8] * nb_[2];
            float aa = fabsf(va[0] * ax + va[1] * ay + va[2] * az);
            float ab = fabsf(va[0] * bx + va[1] * by + va[2] * bz);
            float ba = fabsf(vb[0] * ax + vb[1] * ay + vb[2] * az);
            float bb = fabsf(vb[0] * bx + vb[1] * by + vb[2] * bz);
            a_nb += aa + ab + ba + bb - 2.0f;
        }
    }
    __shared__ float smem[NWAVES];
    int lane = threadIdx.x & 31, wave = threadIdx.x >> 5;
    float t0 = wave_sum_wmma(a_nb);
    if (lane == 0) smem[wave] = t0;
    __syncthreads();
    if (threadIdx.x == 0) {
        float s = 0.f;
        for (int w = 0; w < NWAVES; ++w) s += smem[w];
        atomicAdd(&acc[5], s);
    }
}

__global__ void finalize_kernel(const float* __restrict__ acc,
                                float* __restrict__ out, int N, int M, int K) {
    if (threadIdx.x != 0 || blockIdx.x != 0) return;
    float invN  = 1.0f / (float)N;
    float invM  = 1.0f / (float)M;
    float inv3N = 1.0f / (3.0f * (float)N);

    float sdf    = acc[0] * invN;
    float inter  = acc[1] * invM;
    float eik    = acc[2] / ((float)M + (float)N);
    float morse  = 0.5f * acc[3] * inv3N;
    float thhess = 0.5f * acc[4] * inv3N;
    float thnb   = acc[5] / ((float)N * (float)K);

    float loss = 7000.0f * sdf + 600.0f * inter + 50.0f * eik
               + 3.0f * morse + 10.0f * thhess + 30.0f * thnb;
    out[0] = loss;
    out[1] = sdf;
    out[2] = inter;
    out[3] = eik;
    out[4] = morse;
    out[5] = thhess;
    out[6] = thnb;
}

extern "C" void kernel_launch(void* const* d_in, const int* in_sizes, int n_in,
                              void* d_out, int out_size, void* d_ws, size_t ws_size,
                              hipStream_t stream) {
    const float* manifold_pred    = (const float*)d_in[0];
    const float* nonmanifold_pred = (const float*)d_in[1];
    const float* mnfld_grad       = (const float*)d_in[2];
    const float* nonmnfld_grad    = (const float*)d_in[3];
    const float* mnfld_hessian    = (const float*)d_in[4];
    const float* mnfld_n_gt       = (const float*)d_in[5];
    const float* theta            = (const float*)d_in[6];
    const float* local_u          = (const float*)d_in[7];
    const float* local_v          = (const float*)d_in[8];
    const float* rot              = (const float*)d_in[9];
    const int*   neighbors        = (const int*)d_in[10];

    int N = in_sizes[0];
    int M = in_sizes[1];
    int K = (N > 0) ? (in_sizes[10] / N) : 4;

    float* acc = (float*)d_ws;  // 6 f32 accumulators (zeroed each launch)

    zero_acc_kernel<<<1, 32, 0, stream>>>(acc);

    int gN = (N + BLOCK - 1) / BLOCK;
    int gM = (M + BLOCK - 1) / BLOCK;

    vertex_kernel<<<gN, BLOCK, 0, stream>>>(manifold_pred, mnfld_grad,
                                            mnfld_hessian, mnfld_n_gt, theta,
                                            local_u, local_v, acc, N);
    nonmnfld_kernel<<<gM, BLOCK, 0, stream>>>(nonmanifold_pred, nonmnfld_grad,
                                              acc, M);
    neighbor_kernel<<<gN, BLOCK, 0, stream>>>(theta, local_u, local_v, rot,
                                              neighbors, acc, N, K);
    finalize_kernel<<<1, 1, 0, stream>>>(acc, (float*)d_out, N, M, K);
}